// ScaledDotProdAttn_57827439674168
// MI455X (gfx1250) — compile-verified
//
#include <hip/hip_runtime.h>
#include <hip/hip_bf16.h>

typedef __attribute__((ext_vector_type(2))) float v2f;
typedef __attribute__((ext_vector_type(4))) float v4f;
typedef __attribute__((ext_vector_type(8))) float v8f;

constexpr int kB  = 32;
constexpr int kSQ = 2048;
constexpr int kSK = 2048;
constexpr int kD  = 64;
constexpr float kScale = 0.125f;      // 1/sqrt(64)

constexpr int kWaves   = 4;           // waves per block (wave32)
constexpr int kQT      = 16;          // q rows per wave
constexpr int kKT      = 16;          // k rows per tile
constexpr int kStrK    = 68;          // K-tile row stride: b128-fill aligned, conflict-free b64 frags
constexpr int kStrT    = 18;          // transposed-tile / scratch stride (conflict-free)
constexpr int kNTiles  = kSK / kKT;   // 128

__global__ __launch_bounds__(kWaves * 32)
void sdpa_fused_wmma_f32(const float* __restrict__ q,
                         const float* __restrict__ kmat,
                         const float* __restrict__ vmat,
                         float* __restrict__ out,
                         float* __restrict__ attn)
{
    // K tile row-major [kk][d] (stride 68), V tile transposed [d][kk] (stride 18),
    // per-wave S scratch [m][kk] (stride 18).
    __shared__ __attribute__((aligned(16))) float lds_k [kKT * kStrK];          // 4.25 KB
    __shared__ __attribute__((aligned(16))) float lds_vt[kD * kStrT];           // 4.5 KB
    __shared__ __attribute__((aligned(16))) float lds_s [kWaves * kQT * kStrT]; // 4.5 KB

    const int tid  = threadIdx.x;
    const int wave = tid >> 5;
    const int lane = tid & 31;
    const int n    = lane & 15;        // N / M index within fragment
    const int m    = lane & 15;
    const int hi   = lane >> 4;        // half-wave select
    const int kb   = hi * 2;           // K-pair base within a 4-wide slice

    const int b  = blockIdx.y;
    const int q0 = blockIdx.x * (kWaves * kQT) + wave * kQT;

    // ---- Preload Q fragments (A-matrix layout, ISA 7.12.2): reused for all 128 k-tiles ----
    const float* qrow = q + ((size_t)b * kSQ + (size_t)(q0 + m)) * kD;
    v2f qa[16];
#pragma unroll
    for (int dk = 0; dk < 16; ++dk) {
        qa[dk] = *(const v2f*)(qrow + dk * 4 + kb);   // Q[q0+m][4dk+kb .. +1]
    }

    const v8f vz = {0.f,0.f,0.f,0.f,0.f,0.f,0.f,0.f};
    v8f acc[4] = {vz, vz, vz, vz};                     // O tile 16x64 accumulator

    float* sw = lds_s + wave * (kQT * kStrT);

    // Hoisted, incrementally-advanced pointers (row offsets fold into store IOFFSET).
    const int fkk = (tid & 0xF0) >> 4;                 // fill: tile row for chunk 0 (tid/16)
    const int fd4 = (tid & 15) << 2;                   // fill: d 0..60 step 4
    const float* kg = kmat + ((size_t)b * kSK + fkk) * kD + fd4;
    const float* vg = vmat + ((size_t)b * kSK + fkk) * kD + fd4;
    float* aptr = attn + ((size_t)b * kSQ + q0 + hi * 8) * kSK + n;

#pragma unroll 1
    for (int kt = 0; kt < kNTiles; ++kt) {
        __syncthreads();   // previous iteration done consuming lds_k/lds_vt

        // ---- Cooperative coalesced fill: 16x64 f32 K & V tiles ----
#pragma unroll
        for (int r = 0; r < 2; ++r) {
            const int kk = fkk + r * 8;               // chunk row (128 thr = 8 rows/pass)
            v4f kf = *(const v4f*)(kg + (size_t)r * 8 * kD);
            v4f vf = *(const v4f*)(vg + (size_t)r * 8 * kD);
            *(v4f*)(lds_k + kk * kStrK + fd4) = kf;   // single ds_store_b128
#pragma unroll
            for (int j = 0; j < 4; ++j) {
                lds_vt[(fd4 + j) * kStrT + kk] = vf[j];
            }
            if (kt + 1 < kNTiles) {                   // pull next tile toward L2/L0
                __builtin_prefetch(kg + (size_t)(r * 8 + kKT) * kD, 0, 1);
                __builtin_prefetch(vg + (size_t)(r * 8 + kKT) * kD, 0, 1);
            }
        }
        __syncthreads();

        // ---- S = (Q_tile . K_tile^T): 16 x v_wmma_f32_16x16x4_f32 over D=64 ----
        v8f s = vz;
#pragma unroll
        for (int dk = 0; dk < 16; ++dk) {
            const int d0 = dk * 4 + kb;
            v2f bf = *(const v2f*)(lds_k + n * kStrK + d0);  // K[k0+n][d0..d0+1], one b64
            s = __builtin_amdgcn_wmma_f32_16x16x4_f32(false, qa[dk], false, bf,
                                                      (short)0, s, false, false);
        }

        // ---- Scale, emit attn tile to global, stash S in per-wave LDS scratch ----
#pragma unroll
        for (int r = 0; r < 8; ++r) {
            float sv = s[r] * kScale;
            s[r] = sv;
            aptr[(size_t)r * kSK] = sv;               // C/D layout row r (+hi*8 in base)
            sw[(r + hi * 8) * kStrT + n] = sv;
        }
        asm volatile("s_wait_dscnt 0x0" ::: "memory"); // S scratch visible to own wave reads

        // ---- O += S . V_tile: 4 K-slices x 4 N-blocks of v_wmma_f32_16x16x4_f32 ----
#pragma unroll
        for (int ks = 0; ks < 4; ++ks) {
            const int base = ks * 4 + kb;
            v2f af = *(const v2f*)(sw + m * kStrT + base);   // S[m][base .. base+1]
#pragma unroll
            for (int nb = 0; nb < 4; ++nb) {
                v2f bf = *(const v2f*)(lds_vt + (nb * 16 + n) * kStrT + base); // V[base..+1][col]
                acc[nb] = __builtin_amdgcn_wmma_f32_16x16x4_f32(false, af, false, bf,
                                                                (short)0, acc[nb], false, false);
            }
        }

        // advance incrementally: K/V tile +16 rows, attn tile +16 cols
        kg += (size_t)kKT * kD;
        vg += (size_t)kKT * kD;
        aptr += kKT;
    }

    // ---- Write O tile: out[b][q0 + row][col] ----
    float* orow = out + ((size_t)b * kSQ + q0 + hi * 8) * kD;
#pragma unroll
    for (int nb = 0; nb < 4; ++nb) {
#pragma unroll
        for (int r = 0; r < 8; ++r) {
            orow[(size_t)r * kD + nb * 16 + n] = acc[nb][r];
        }
    }
}

extern "C" void kernel_launch(void* const* d_in, const int* in_sizes, int n_in,
                              void* d_out, int out_size, void* d_ws, size_t ws_size,
                              hipStream_t stream) {
    (void)in_sizes; (void)n_in; (void)d_ws; (void)ws_size; (void)out_size;
    const float* q = (const float*)d_in[0];
    const float* k = (const float*)d_in[1];
    const float* v = (const float*)d_in[2];
    float* out  = (float*)d_out;                         // [B, SQ, D] first (return order)
    float* attn = out + (size_t)kB * kSQ * kD;           // then [B, SQ, SK]

    dim3 grid(kSQ / (kWaves * kQT), kB);                 // 32 x 32 blocks
    dim3 block(kWaves * 32);                             // 4 waves of 32
    sdpa_fused_wmma_f32<<<grid, block, 0, stream>>>(q, k, v, out, attn);
}